// MuellerMatrixModel_23802708754563
// MI455X (gfx1250) — compile-verified
//
#include <hip/hip_runtime.h>

#define NB 2
#define HH 512
#define WW 512
#define PLANE (HH*WW)
#define NCH_IN 48
#define NCH_OUT 20
#define EPSF 1e-6f
#define PIF 3.14159265358979323846f

typedef __attribute__((ext_vector_type(2))) float v2f;
typedef __attribute__((ext_vector_type(8))) float v8f;

// ---------------- small linear algebra helpers (per-lane scalar) -------------

__device__ __forceinline__ void inv4(const float* m, float* inv) {
    float i0 =  m[5]*m[10]*m[15] - m[5]*m[11]*m[14] - m[9]*m[6]*m[15] + m[9]*m[7]*m[14] + m[13]*m[6]*m[11] - m[13]*m[7]*m[10];
    float i4 = -m[4]*m[10]*m[15] + m[4]*m[11]*m[14] + m[8]*m[6]*m[15] - m[8]*m[7]*m[14] - m[12]*m[6]*m[11] + m[12]*m[7]*m[10];
    float i8 =  m[4]*m[9]*m[15]  - m[4]*m[11]*m[13] - m[8]*m[5]*m[15] + m[8]*m[7]*m[13] + m[12]*m[5]*m[11] - m[12]*m[7]*m[9];
    float i12= -m[4]*m[9]*m[14]  + m[4]*m[10]*m[13] + m[8]*m[5]*m[14] - m[8]*m[6]*m[13] - m[12]*m[5]*m[10] + m[12]*m[6]*m[9];
    float i1 = -m[1]*m[10]*m[15] + m[1]*m[11]*m[14] + m[9]*m[2]*m[15] - m[9]*m[3]*m[14] - m[13]*m[2]*m[11] + m[13]*m[3]*m[10];
    float i5 =  m[0]*m[10]*m[15] - m[0]*m[11]*m[14] - m[8]*m[2]*m[15] + m[8]*m[3]*m[14] + m[12]*m[2]*m[11] - m[12]*m[3]*m[10];
    float i9 = -m[0]*m[9]*m[15]  + m[0]*m[11]*m[13] + m[8]*m[1]*m[15] - m[8]*m[3]*m[13] - m[12]*m[1]*m[11] + m[12]*m[3]*m[9];
    float i13=  m[0]*m[9]*m[14]  - m[0]*m[10]*m[13] - m[8]*m[1]*m[14] + m[8]*m[2]*m[13] + m[12]*m[1]*m[10] - m[12]*m[2]*m[9];
    float i2 =  m[1]*m[6]*m[15]  - m[1]*m[7]*m[14]  - m[5]*m[2]*m[15] + m[5]*m[3]*m[14] + m[13]*m[2]*m[7]  - m[13]*m[3]*m[6];
    float i6 = -m[0]*m[6]*m[15]  + m[0]*m[7]*m[14]  + m[4]*m[2]*m[15] - m[4]*m[3]*m[14] - m[12]*m[2]*m[7]  + m[12]*m[3]*m[6];
    float i10=  m[0]*m[5]*m[15]  - m[0]*m[7]*m[13]  - m[4]*m[1]*m[15] + m[4]*m[3]*m[13] + m[12]*m[1]*m[7]  - m[12]*m[3]*m[5];
    float i14= -m[0]*m[5]*m[14]  + m[0]*m[6]*m[13]  + m[4]*m[1]*m[14] - m[4]*m[2]*m[13] - m[12]*m[1]*m[6]  + m[12]*m[2]*m[5];
    float i3 = -m[1]*m[6]*m[11]  + m[1]*m[7]*m[10]  + m[5]*m[2]*m[11] - m[5]*m[3]*m[10] - m[9]*m[2]*m[7]   + m[9]*m[3]*m[6];
    float i7 =  m[0]*m[6]*m[11]  - m[0]*m[7]*m[10]  - m[4]*m[2]*m[11] + m[4]*m[3]*m[10] + m[8]*m[2]*m[7]   - m[8]*m[3]*m[6];
    float i11= -m[0]*m[5]*m[11]  + m[0]*m[7]*m[9]   + m[4]*m[1]*m[11] - m[4]*m[3]*m[9]  - m[8]*m[1]*m[7]   + m[8]*m[3]*m[5];
    float i15=  m[0]*m[5]*m[10]  - m[0]*m[6]*m[9]   - m[4]*m[1]*m[10] + m[4]*m[2]*m[9]  + m[8]*m[1]*m[6]   - m[8]*m[2]*m[5];
    float det = m[0]*i0 + m[1]*i4 + m[2]*i8 + m[3]*i12;
    float r = 1.0f/det;
    inv[0]=i0*r;  inv[1]=i1*r;  inv[2]=i2*r;  inv[3]=i3*r;
    inv[4]=i4*r;  inv[5]=i5*r;  inv[6]=i6*r;  inv[7]=i7*r;
    inv[8]=i8*r;  inv[9]=i9*r;  inv[10]=i10*r; inv[11]=i11*r;
    inv[12]=i12*r; inv[13]=i13*r; inv[14]=i14*r; inv[15]=i15*r;
}

__device__ __forceinline__ void mat4mul(const float* a, const float* b, float* c) {
    #pragma unroll
    for (int i = 0; i < 4; ++i)
        #pragma unroll
        for (int j = 0; j < 4; ++j) {
            float s = 0.f;
            #pragma unroll
            for (int k = 0; k < 4; ++k) s += a[i*4+k]*b[k*4+j];
            c[i*4+j] = s;
        }
}

__device__ __forceinline__ float det3(const float* a) {
    return a[0]*(a[4]*a[8]-a[5]*a[7]) - a[1]*(a[3]*a[8]-a[5]*a[6]) + a[2]*(a[3]*a[7]-a[4]*a[6]);
}

__device__ __forceinline__ void inv3(const float* a, float* o) {
    float c00 =  a[4]*a[8]-a[5]*a[7];
    float c01 = -(a[3]*a[8]-a[5]*a[6]);
    float c02 =  a[3]*a[7]-a[4]*a[6];
    float det = a[0]*c00 + a[1]*c01 + a[2]*c02;
    float r = 1.0f/det;
    o[0]=c00*r; o[1]=(a[2]*a[7]-a[1]*a[8])*r; o[2]=(a[1]*a[5]-a[2]*a[4])*r;
    o[3]=c01*r; o[4]=(a[0]*a[8]-a[2]*a[6])*r; o[5]=(a[2]*a[3]-a[0]*a[5])*r;
    o[6]=c02*r; o[7]=(a[1]*a[6]-a[0]*a[7])*r; o[8]=(a[0]*a[4]-a[1]*a[3])*r;
}

__device__ __forceinline__ void mat3mul(const float* a, const float* b, float* c) {
    #pragma unroll
    for (int i = 0; i < 3; ++i)
        #pragma unroll
        for (int j = 0; j < 3; ++j) {
            float s = 0.f;
            #pragma unroll
            for (int k = 0; k < 3; ++k) s += a[i*3+k]*b[k*3+j];
            c[i*3+j] = s;
        }
}

// closed-form symmetric 3x3 eigenvalues (any order; callers use symmetric fns)
__device__ __forceinline__ void eig3sym(const float* A, float& l0, float& l1, float& l2) {
    float p1 = A[1]*A[1] + A[2]*A[2] + A[5]*A[5];
    float q  = (A[0]+A[4]+A[8]) * (1.f/3.f);
    float b00 = A[0]-q, b11 = A[4]-q, b22 = A[8]-q;
    float p2 = b00*b00 + b11*b11 + b22*b22 + 2.f*p1;
    float p  = sqrtf(fmaxf(p2, 1e-30f) * (1.f/6.f));
    float ip = 1.f/p;
    float B0=b00*ip, B1=A[1]*ip, B2=A[2]*ip, B4=b11*ip, B5=A[5]*ip, B8=b22*ip;
    float detB = B0*(B4*B8 - B5*B5) - B1*(B1*B8 - B5*B2) + B2*(B1*B5 - B4*B2);
    float r = fminf(fmaxf(0.5f*detB, -1.f), 1.f);
    float phi = acosf(r) * (1.f/3.f);
    l2 = q + 2.f*p*cosf(phi);
    l0 = q + 2.f*p*cosf(phi + 2.0943951023931953f);
    l1 = 3.f*q - l0 - l2;
}

// Pauli matrices (constant-folds under full unroll)
__device__ __forceinline__ float Sre(int i, int a, int b) {
    const float t[4][2][2] = {{{1,0},{0,1}},{{1,0},{0,-1}},{{0,1},{1,0}},{{0,0},{0,0}}};
    return t[i][a][b];
}
__device__ __forceinline__ float Sim(int i, int a, int b) {
    const float t[4][2][2] = {{{0,0},{0,0}},{{0,0},{0,0}},{{0,0},{0,0}},{{0,-1},{1,0}}};
    return t[i][a][b];
}

// ---------------- WMMA batched 4x4 matmul: 4 pixels per v_wmma ---------------
// D(16x16) = A(16x4) x B(4x16); A stacks 4 pixels' 4x4 vertically,
// B = [B0|B1|B2|B3] horizontally; diagonal 4x4 blocks are the products.
// Scatter uses an arithmetic blend (no dynamic vector extract) and writes
// unconditionally: inactive lanes target a never-read LDS trash row, so EXEC
// stays all-ones across the whole loop.
__device__ __forceinline__ void wave_mm4(const float (*As)[16], const float (*Bs)[16],
                                         float (*Ds)[16], float* trash) {
    const int tid   = threadIdx.x;
    const int lane  = tid & 31;
    const int wave  = tid >> 5;
    const int lhalf = lane >> 4;     // 0: K=0,1 rows 0-7 ; 1: K=2,3 rows 8-15
    const int l15   = lane & 15;
    const int pp    = l15 >> 2;      // pixel-in-group (== column-block for B)
    const int ii    = l15 & 3;       // row in 4x4 for A / col in block for B
    const int k0    = 2*lhalf;
    const bool act = (lhalf == 0) ? (pp < 2) : (pp >= 2);
    const bool hi  = (lhalf == 0) ? (pp == 1) : (pp == 3);
    const float h  = hi ? 1.f : 0.f;   // select d[4..7] vs d[0..3] via fma blend
    const v8f cz = {};
    #pragma unroll
    for (int g = 0; g < 8; ++g) {
        const int base = wave*32 + g*4;
        v2f a, bv;
        a.x  = As[base+pp][ii*4 + k0];
        a.y  = As[base+pp][ii*4 + k0 + 1];
        bv.x = Bs[base+pp][(k0  )*4 + ii];
        bv.y = Bs[base+pp][(k0+1)*4 + ii];
        v8f d = __builtin_amdgcn_wmma_f32_16x16x4_f32(
            /*neg_a=*/false, a, /*neg_b=*/false, bv,
            /*c_mod=*/(short)0, cz, /*reuse_a=*/false, /*reuse_b=*/false);
        float* dst = act ? &Ds[base+pp][ii] : &trash[ii];
        dst[0]  = fmaf(h, d[4] - d[0], d[0]);
        dst[4]  = fmaf(h, d[5] - d[1], d[1]);
        dst[8]  = fmaf(h, d[6] - d[2], d[2]);
        dst[12] = fmaf(h, d[7] - d[3], d[3]);
    }
}

// ---------------- main per-pixel kernel --------------------------------------

__global__ __launch_bounds__(256) void mueller_main(const float* __restrict__ x,
                                                    float* __restrict__ out) {
    __shared__ float sA[256][16];
    __shared__ float sB[256][16];
    __shared__ float sD[256][16];
    __shared__ float sTrash[16];

    const int tid = threadIdx.x;
    const int gid = blockIdx.x * 256 + tid;   // grid sized exactly: no divergence
    const int b   = gid / PLANE;
    const int hw  = gid - b * PLANE;
    const size_t ibase = (size_t)b * NCH_IN * PLANE + hw;

    float F[16], Am[16], Wm[16];
    #pragma unroll
    for (int k = 0; k < 16; ++k) {
        F[k]  = x[ibase + (size_t)(k     ) * PLANE];
        Am[k] = x[ibase + (size_t)(k + 16) * PLANE];
        Wm[k] = x[ibase + (size_t)(k + 32) * PLANE];
    }

    float inten = 0.f;
    #pragma unroll
    for (int k = 0; k < 16; ++k) inten += F[k];
    inten *= (1.f/16.f);

    float Ai[16], Wi[16];
    inv4(Am, Ai);
    inv4(Wm, Wi);

    // ---- M = Ai @ F @ Wi via WMMA (4 pixels per instruction, LDS staged) ----
    #pragma unroll
    for (int k = 0; k < 16; ++k) { sA[tid][k] = Ai[k]; sB[tid][k] = F[k]; }
    __syncthreads();
    wave_mm4(sA, sB, sD, sTrash);         // T = Ai*F
    __syncthreads();
    #pragma unroll
    for (int k = 0; k < 16; ++k) sB[tid][k] = Wi[k];
    __syncthreads();
    wave_mm4(sD, sB, sA, sTrash);         // M = T*Wi
    __syncthreads();

    float M[16];
    #pragma unroll
    for (int k = 0; k < 16; ++k) M[k] = sA[tid][k];
    float r00 = 1.f / M[0];
    #pragma unroll
    for (int k = 0; k < 16; ++k) M[k] *= r00;

    // ---- charpoly mask: Hc = 0.25 * sum_ij M[i][j] (S_i kron conj(S_j)) ----
    float Hre[16], Him[16];
    #pragma unroll
    for (int row = 0; row < 4; ++row) {
        #pragma unroll
        for (int col = 0; col < 4; ++col) {
            const int a = row >> 1, c = row & 1, bb = col >> 1, dd = col & 1;
            float sre = 0.f, sim = 0.f;
            #pragma unroll
            for (int i = 0; i < 4; ++i) {
                const float ur = Sre(i, a, bb), ui = Sim(i, a, bb);
                #pragma unroll
                for (int j = 0; j < 4; ++j) {
                    const float wr = Sre(j, c, dd), wi = -Sim(j, c, dd);
                    const float mij = M[i*4+j];
                    sre += mij * (ur*wr - ui*wi);
                    sim += mij * (ur*wi + ui*wr);
                }
            }
            Hre[row*4+col] = 0.25f * sre;
            Him[row*4+col] = 0.25f * sim;
        }
    }
    float H2re[16], H2im[16];
    #pragma unroll
    for (int i = 0; i < 4; ++i)
        #pragma unroll
        for (int j = 0; j < 4; ++j) {
            float re = 0.f, im = 0.f;
            #pragma unroll
            for (int k = 0; k < 4; ++k) {
                re += Hre[i*4+k]*Hre[k*4+j] - Him[i*4+k]*Him[k*4+j];
                im += Hre[i*4+k]*Him[k*4+j] + Him[i*4+k]*Hre[k*4+j];
            }
            H2re[i*4+j] = re; H2im[i*4+j] = im;
        }
    const float t1 = Hre[0]+Hre[5]+Hre[10]+Hre[15];
    const float t2 = H2re[0]+H2re[5]+H2re[10]+H2re[15];
    float t3 = 0.f, t4 = 0.f;
    #pragma unroll
    for (int i = 0; i < 4; ++i)
        #pragma unroll
        for (int j = 0; j < 4; ++j) {
            t3 += H2re[i*4+j]*Hre[j*4+i]  - H2im[i*4+j]*Him[j*4+i];
            t4 += H2re[i*4+j]*H2re[j*4+i] - H2im[i*4+j]*H2im[j*4+i];
        }
    const float c2 = 0.5f*(t1*t1 - t2);
    const float c3 = (t1*t1*t1 - 3.f*t1*t2 + 2.f*t3) * (1.f/6.f);
    const float c4 = (t1*t1*t1*t1 - 6.f*t1*t1*t2 + 3.f*t2*t2 + 8.f*t1*t3 - 6.f*t4) * (1.f/24.f);
    const bool vmask = (t1 >= -EPSF) && (c2 >= -EPSF) && (c3 >= -EPSF) && (c4 >= -EPSF);

    // ---- Lu-Chipman (only mR off-diagonals needed for azimuth) -------------
    float Mu[16];
    #pragma unroll
    for (int k = 0; k < 16; ++k) {
        const float id = ((k & 3) == (k >> 2)) ? 1.f : 0.f;
        Mu[k] = vmask ? M[k] : id;
    }
    const float D0 = Mu[1], D1 = Mu[2], D2 = Mu[3];
    float d2 = fminf(fmaxf(D0*D0 + D1*D1 + D2*D2, 0.f), 1.f - EPSF);
    const float dd = sqrtf(d2);
    const float invd = 1.f / fmaxf(dd, EPSF);
    const float h0 = D0*invd, h1 = D1*invd, h2 = D2*invd;
    const float sq = sqrtf(1.f - d2);
    const float om = 1.f - sq;
    float mD[9] = { sq + om*h0*h0, om*h0*h1,      om*h0*h2,
                    om*h1*h0,      sq + om*h1*h1, om*h1*h2,
                    om*h2*h0,      om*h2*h1,      sq + om*h2*h2 };
    float MD[16] = { 1.f, D0, D1, D2,
                     D0, mD[0], mD[1], mD[2],
                     D1, mD[3], mD[4], mD[5],
                     D2, mD[6], mD[7], mD[8] };
    float MDi[16]; inv4(MD, MDi);
    float Mp[16];  mat4mul(Mu, MDi, Mp);
    float mp[9] = { Mp[5], Mp[6], Mp[7],  Mp[9], Mp[10], Mp[11],  Mp[13], Mp[14], Mp[15] };
    float mmT[9];
    #pragma unroll
    for (int i = 0; i < 3; ++i)
        #pragma unroll
        for (int j = 0; j < 3; ++j) {
            float s = 0.f;
            #pragma unroll
            for (int k = 0; k < 3; ++k) s += mp[i*3+k]*mp[j*3+k];
            mmT[i*3+j] = s;
        }
    float l0, l1, l2;
    eig3sym(mmT, l0, l1, l2);
    const float s1 = sqrtf(fmaxf(l0, 0.f));
    const float s2 = sqrtf(fmaxf(l1, 0.f));
    const float s3 = sqrtf(fmaxf(l2, 0.f));
    const float sgn = (det3(mp) < 0.f) ? -1.f : 1.f;
    const float e2s = s1*s2 + s2*s3 + s3*s1 + EPSF;
    const float ssum = s1 + s2 + s3;
    const float sprod = s1*s2*s3;
    float A3[9], Bm3[9];
    #pragma unroll
    for (int k = 0; k < 9; ++k) { A3[k] = mmT[k]; Bm3[k] = ssum*mmT[k]; }
    A3[0] += e2s; A3[4] += e2s; A3[8] += e2s;
    Bm3[0] += sprod; Bm3[4] += sprod; Bm3[8] += sprod;
    float A3i[9]; inv3(A3, A3i);
    float mDel[9]; mat3mul(A3i, Bm3, mDel);
    #pragma unroll
    for (int k = 0; k < 9; ++k) mDel[k] *= sgn;
    float C3[9];
    #pragma unroll
    for (int k = 0; k < 9; ++k) C3[k] = mDel[k];
    C3[0] += EPSF; C3[4] += EPSF; C3[8] += EPSF;
    float C3i[9]; inv3(C3, C3i);
    float mR[9]; mat3mul(C3i, mp, mR);

    // azimuth = mod(0.5*atan2(a2, a1), pi) * 180/pi ; positive 2*sinR cancels
    const float a1n = mR[5] - mR[7];    // mR[1][2] - mR[2][1]
    const float a2n = mR[6] - mR[2];    // mR[2][0] - mR[0][2]
    float azim = fmodf(0.5f * atan2f(a2n, a1n), PIF);
    if (azim < 0.f) azim += PIF;
    azim *= (180.f / PIF);

    // ---- outputs: [intensity(1), M(16), azi(1), std(1 by k2), v(1)] --------
    const size_t obase = (size_t)b * NCH_OUT * PLANE + hw;
    out[obase] = inten;
    #pragma unroll
    for (int k = 0; k < 16; ++k) out[obase + (size_t)(1+k)*PLANE] = M[k];
    out[obase + (size_t)17*PLANE] = azim;
    out[obase + (size_t)19*PLANE] = vmask ? 1.f : 0.f;
}

// ---------------- rolling circular std (4x4 edge-clamped window) -------------

__global__ __launch_bounds__(256) void circstd_kernel(float* __restrict__ out) {
    const int gid = blockIdx.x * 256 + threadIdx.x;
    const int b   = gid / PLANE;
    const int hw  = gid - b * PLANE;
    const int h   = hw >> 9;
    const int w   = hw & (WW - 1);
    const float* azi = out + ((size_t)b * NCH_OUT + 17) * PLANE;
    float C = 0.f, S = 0.f;
    #pragma unroll
    for (int di = -1; di <= 2; ++di) {
        const int hh = min(max(h + di, 0), HH - 1);
        #pragma unroll
        for (int dj = -1; dj <= 2; ++dj) {
            const int wc = min(max(w + dj, 0), WW - 1);
            const float ang = azi[hh * WW + wc] * 0.03490658503988659f; // 2*pi/180
            C += cosf(ang);
            S += sinf(ang);
        }
    }
    C *= (1.f/16.f); S *= (1.f/16.f);
    float Rm = fminf(fmaxf(sqrtf(C*C + S*S + EPSF), EPSF), 1.f - 1e-7f);
    out[((size_t)b * NCH_OUT + 18) * PLANE + hw] =
        sqrtf(-2.f * logf(Rm)) * 28.64788975654116f; // 0.5/pi*180
}

// ---------------- launcher ---------------------------------------------------

extern "C" void kernel_launch(void* const* d_in, const int* in_sizes, int n_in,
                              void* d_out, int out_size, void* d_ws, size_t ws_size,
                              hipStream_t stream) {
    (void)in_sizes; (void)n_in; (void)out_size; (void)d_ws; (void)ws_size;
    const float* x = (const float*)d_in[0];
    float* out = (float*)d_out;
    const int total = NB * PLANE;         // 524288, divisible by 256
    dim3 blk(256), grd(total / 256);
    mueller_main<<<grd, blk, 0, stream>>>(x, out);
    circstd_kernel<<<grd, blk, 0, stream>>>(out);
}